// Arty_17454747091183
// MI455X (gfx1250) — compile-verified
//
#include <hip/hip_runtime.h>

// Fully-fused tiny-MLP forward for gfx1250 (MI455X).
//  - Hidden 16x16 layers run as V_WMMA_F32_16X16X4_F32 chains (4 per layer),
//    computing h^T_{l+1} = W_l * h^T_l with W as the A operand so that the
//    D->B inter-layer conversion is just a lane(n)<->lane(n+16) half-swap.
//  - The half-swap is v_permlanex16_b32 (pure VALU, no LDS pipe, no dscnt
//    waits on the serial layer chain); ds_swizzle fallback if unavailable.
//  - All weights live in VGPRs per wave, amortized over many 16-point tiles.
//  - tanh/sigmoid via v_exp_f32 + v_rcp_f32 (inf-correct saturation).

typedef __attribute__((ext_vector_type(2))) float v2f;
typedef __attribute__((ext_vector_type(8))) float v8f;

#if __has_builtin(__builtin_amdgcn_permlanex16)
// Cross-half swap: identity lane selects -> lane i of each half reads lane i
// of the opposite half (== swap across lane 16 boundary). VALU-only.
__device__ __forceinline__ float swapx16(float v) {
    int i = __float_as_int(v);
    int r = __builtin_amdgcn_permlanex16(i, i, 0x76543210, 0xFEDCBA98, false, false);
    return __int_as_float(r);
}
#else
// ds_swizzle group-of-32 encoding: offset[14:10]=xor, [9:5]=or, [4:0]=and
// SWAPX16: xor=0x10, or=0, and=0x1f -> 0x401f
__device__ __forceinline__ float swapx16(float v) {
    return __int_as_float(__builtin_amdgcn_ds_swizzle(__float_as_int(v), 0x401f));
}
#endif

__device__ __forceinline__ float fast_tanh(float x) {
    // tanh(x) = 1 - 2/(exp(2x)+1); exp(2x) = exp2(x * 2*log2(e))
    float e = __builtin_amdgcn_exp2f(x * 2.885390081777927f);
    float r = __builtin_amdgcn_rcpf(e + 1.0f);
    return fmaf(-2.0f, r, 1.0f);
}

__device__ __forceinline__ float fast_sigmoid(float x) {
    float e = __builtin_amdgcn_exp2f(x * -1.4426950408889634f);
    return __builtin_amdgcn_rcpf(1.0f + e);
}

__global__ void __launch_bounds__(256) mlp_fused_kernel(
    const float* __restrict__ x,     // [N,2]
    const float* __restrict__ W0,    // [16,2]
    const float* __restrict__ b0,    // [16]
    const float* __restrict__ Wh,    // [8,16,16]
    const float* __restrict__ Wout,  // [3,16]
    float* __restrict__ out,         // [N,3]
    int nTiles)
{
    const int lane  = threadIdx.x & 31;
    const int wave  = threadIdx.x >> 5;
    const int gwave = blockIdx.x * (blockDim.x >> 5) + wave;
    const int nwav  = gridDim.x * (blockDim.x >> 5);

    const int  m    = lane & 15;        // point-in-tile / matrix-row index
    const bool hi   = lane >= 16;       // upper half-wave
    const int  fb   = hi ? 8 : 0;       // feature base held by this lane (C/D layout)
    const int  koff = hi ? 2 : 0;       // K offset for A/B f32 fragments

    // ---------------- per-wave register-resident weights ----------------
    // Layer 0: lane handles output rows fb..fb+7
    float w0x[8], w0y[8], bb[8];
#pragma unroll
    for (int r = 0; r < 8; ++r) {
        w0x[r] = W0[(fb + r) * 2 + 0];
        w0y[r] = W0[(fb + r) * 2 + 1];
        bb[r]  = b0[fb + r];
    }
    // Hidden layers as WMMA A-fragments (16x4 f32): layer l, K-chunk c.
    // A layout: lanes 0-15 -> (M=lane, K=4c..4c+1), lanes 16-31 -> (M=lane-16, K=4c+2..4c+3)
    v2f wa[8][4];
#pragma unroll
    for (int l = 0; l < 8; ++l)
#pragma unroll
        for (int c = 0; c < 4; ++c) {
            const float* p = Wh + l * 256 + m * 16 + 4 * c + koff;
            wa[l][c].x = p[0];
            wa[l][c].y = p[1];
        }
    // Output layer: lane holds input features fb..fb+7 of each of 3 rows
    float wo[3][8];
#pragma unroll
    for (int j = 0; j < 3; ++j)
#pragma unroll
        for (int r = 0; r < 8; ++r)
            wo[j][r] = Wout[j * 16 + fb + r];

    // ---------------- stream tiles of 16 points ----------------
    for (int t = gwave; t < nTiles; t += nwav) {
        const int base = t * 16;
        // both half-waves fetch the same point (same cacheline, coalesced)
        const float* xp = x + (size_t)(base + m) * 2;
        float x0 = __builtin_nontemporal_load(xp + 0);
        float x1 = __builtin_nontemporal_load(xp + 1);

        // Layer 0 (2->16 + bias + tanh) in VALU, result directly in D-layout:
        // lane n<16 : d[r] = h[point n][feature r] ; lane n+16 : feature 8+r
        float d[8];
#pragma unroll
        for (int r = 0; r < 8; ++r)
            d[r] = fast_tanh(fmaf(x0, w0x[r], fmaf(x1, w0y[r], bb[r])));

        // 8 hidden layers: h^T <- tanh(W * h^T), W as WMMA-A, h^T as WMMA-B
#pragma unroll
        for (int l = 0; l < 8; ++l) {
            float s[8];
#pragma unroll
            for (int r = 0; r < 8; ++r) s[r] = swapx16(d[r]);
            // Assemble B fragments (4x16 f32) for the 4 K-chunks:
            v2f bf[4];
            bf[0].x = hi ? s[2] : d[0];  bf[0].y = hi ? s[3] : d[1];
            bf[1].x = hi ? s[6] : d[4];  bf[1].y = hi ? s[7] : d[5];
            bf[2].x = hi ? d[2] : s[0];  bf[2].y = hi ? d[3] : s[1];
            bf[3].x = hi ? d[6] : s[4];  bf[3].y = hi ? d[7] : s[5];

            v8f acc = {};
            acc = __builtin_amdgcn_wmma_f32_16x16x4_f32(false, wa[l][0], false, bf[0], (short)0, acc, false, false);
            acc = __builtin_amdgcn_wmma_f32_16x16x4_f32(false, wa[l][1], false, bf[1], (short)0, acc, false, false);
            acc = __builtin_amdgcn_wmma_f32_16x16x4_f32(false, wa[l][2], false, bf[2], (short)0, acc, false, false);
            acc = __builtin_amdgcn_wmma_f32_16x16x4_f32(false, wa[l][3], false, bf[3], (short)0, acc, false, false);
#pragma unroll
            for (int r = 0; r < 8; ++r) d[r] = fast_tanh(acc[r]);
        }

        // Output layer 16->3 + sigmoid: per-lane partial dot over 8 features,
        // reduce across the lane pair with one cross-half swap.
        float o[3];
#pragma unroll
        for (int j = 0; j < 3; ++j) {
            float p = 0.0f;
#pragma unroll
            for (int r = 0; r < 8; ++r) p = fmaf(d[r], wo[j][r], p);
            o[j] = p + swapx16(p);
        }
        if (!hi) {
            float* q = out + (size_t)(base + m) * 3;
            __builtin_nontemporal_store(fast_sigmoid(o[0]), q + 0);
            __builtin_nontemporal_store(fast_sigmoid(o[1]), q + 1);
            __builtin_nontemporal_store(fast_sigmoid(o[2]), q + 2);
        }
    }
}

extern "C" void kernel_launch(void* const* d_in, const int* in_sizes, int n_in,
                              void* d_out, int out_size, void* d_ws, size_t ws_size,
                              hipStream_t stream) {
    const float* x    = (const float*)d_in[0];  // [N,2]
    const float* W0   = (const float*)d_in[1];  // [16,2]
    const float* b0   = (const float*)d_in[2];  // [16]
    const float* Wh   = (const float*)d_in[3];  // [8,16,16]
    const float* Wout = (const float*)d_in[4];  // [3,16]
    float* out = (float*)d_out;

    const int N      = in_sizes[0] / 2;   // number of points
    const int nTiles = N / 16;            // 16 points per WMMA tile

    const int block = 256;                // 8 wave32 per block
    int grid = 4096;                      // ~8 tiles per wave at N=4.19M
    int maxGrid = (nTiles + (block / 32) - 1) / (block / 32);
    if (grid > maxGrid) grid = maxGrid;

    mlp_fused_kernel<<<grid, block, 0, stream>>>(x, W0, b0, Wh, Wout, out, nTiles);
}